// DecoderAttention_29953101922974
// MI455X (gfx1250) — compile-verified
//
#include <hip/hip_runtime.h>

// ---------------- problem constants ----------------
#define S_LEN   64
#define E_DIM   512
#define Z_DIM   512
#define H_DIM   1024
#define G4      4096           // 4*H
#define V_DIM   32000
#define LENC    64

// ---------------- ws layout (floats) ----------------
#define OFF_G0    0                       // 64*4096 = 262144
#define OFF_OUTS  262144                  // 64*1024 = 65536
#define OFF_H0B   327680                  // 2*1024 double buffer
#define OFF_H1B   329728                  // 2*1024
#define OFF_C0    331776                  // 1024
#define OFF_C1    332800                  // 1024
#define OFF_CNT   333824                  // 256 unsigned counters

// ---------------- out layout (floats) ----------------
#define OUT_LOGITS 0                      // 64*32000
#define OUT_ATTN   2048000                // 64*2048
#define OUT_HT     2179072                // 2*1024
#define OUT_CT     2181120                // 2*1024

// padded row pitch for the outs tile staged in LDS (1024 + 4 -> bank offset 4/row)
#define ROWP      1028
#define SMEM_BYTES (64 * ROWP * 4)        // 263168 B  (<= 320 KB WGP LDS)

typedef __attribute__((ext_vector_type(2))) float v2f;
typedef __attribute__((ext_vector_type(8))) float v8f;

__device__ __forceinline__ float sigm(float x) { return 1.0f / (1.0f + __expf(-x)); }

__device__ __forceinline__ void grid_barrier(unsigned* cnt, unsigned nblocks) {
    __threadfence();
    __syncthreads();
    if (threadIdx.x == 0) {
        __hip_atomic_fetch_add(cnt, 1u, __ATOMIC_ACQ_REL, __HIP_MEMORY_SCOPE_AGENT);
        while (__hip_atomic_load(cnt, __ATOMIC_ACQUIRE, __HIP_MEMORY_SCOPE_AGENT) < nblocks) {
            __builtin_amdgcn_s_sleep(1);
        }
    }
    __syncthreads();
}

// ============================================================
// Kernel 0: init state (h/c from inputs, barrier counters = 0)
// ============================================================
__global__ __launch_bounds__(1024) void init_state(const float* __restrict__ h0_in,
                                                   const float* __restrict__ c0_in,
                                                   float* __restrict__ ws) {
    int i = threadIdx.x;
    if (i < H_DIM) {
        ws[OFF_H0B + i] = h0_in[i];              // layer0 h, buffer 0
        ws[OFF_H1B + i] = h0_in[H_DIM + i];      // layer1 h, buffer 0
        ws[OFF_C0 + i]  = c0_in[i];
        ws[OFF_C1 + i]  = c0_in[H_DIM + i];
    }
    if (i < 256) ((unsigned*)(ws + OFF_CNT))[i] = 0u;
}

// ============================================================
// Kernel 1: G0 = X @ W_ih0^T + b_ih0 + b_hh0
//   X = [dec | z_broadcast].  The z half is identical for all 64
//   rows, so it is folded into the per-column bias with a scalar
//   dot; WMMA runs only over the dec half (K=512), branch-free.
//   one wave per 16-wide N tile; covers all 4 M tiles.
//   grid: 256 waves -> 32 blocks x 256
// ============================================================
__global__ __launch_bounds__(256) void g0_wmma(const float* __restrict__ dec,
                                               const float* __restrict__ z,
                                               const float* __restrict__ Wih0,
                                               const float* __restrict__ bih0,
                                               const float* __restrict__ bhh0,
                                               float* __restrict__ g0) {
    const int wave = (blockIdx.x * blockDim.x + threadIdx.x) >> 5;   // 0..255 = N tile
    const int lane = threadIdx.x & 31;
    const int half = lane >> 4;
    const int l15  = lane & 15;
    const int n0   = wave * 16;
    const int col  = n0 + l15;

    // WMMA over the decoder-input half of K
    v8f c0 = {}, c1 = {}, c2 = {}, c3 = {};
    for (int k = 0; k < E_DIM; k += 4) {
        const int kb = k + half * 2;
        v2f b  = *(const v2f*)(Wih0 + (size_t)col * H_DIM + kb);
        v2f a0 = *(const v2f*)(dec + (0 * 16 + l15) * E_DIM + kb);
        v2f a1 = *(const v2f*)(dec + (1 * 16 + l15) * E_DIM + kb);
        v2f a2 = *(const v2f*)(dec + (2 * 16 + l15) * E_DIM + kb);
        v2f a3 = *(const v2f*)(dec + (3 * 16 + l15) * E_DIM + kb);
        c0 = __builtin_amdgcn_wmma_f32_16x16x4_f32(false, a0, false, b, (short)0, c0, false, false);
        c1 = __builtin_amdgcn_wmma_f32_16x16x4_f32(false, a1, false, b, (short)0, c1, false, false);
        c2 = __builtin_amdgcn_wmma_f32_16x16x4_f32(false, a2, false, b, (short)0, c2, false, false);
        c3 = __builtin_amdgcn_wmma_f32_16x16x4_f32(false, a3, false, b, (short)0, c3, false, false);
    }

    // z contribution: identical for all rows of this column
    float zacc = 0.f;
    {
        const float4* zr = (const float4*)z;
        const float4* wr = (const float4*)(Wih0 + (size_t)col * H_DIM + E_DIM);
#pragma unroll 8
        for (int kk = 0; kk < Z_DIM / 4; ++kk) {
            float4 a = wr[kk], b = zr[kk];
            zacc += a.x * b.x + a.y * b.y + a.z * b.z + a.w * b.w;
        }
    }
    const float bias = bih0[col] + bhh0[col] + zacc;
#pragma unroll
    for (int r = 0; r < 8; ++r) {
        const int m = r + half * 8;
        g0[(size_t)(0 * 16 + m) * G4 + col] = c0[r] + bias;
        g0[(size_t)(1 * 16 + m) * G4 + col] = c1[r] + bias;
        g0[(size_t)(2 * 16 + m) * G4 + col] = c2[r] + bias;
        g0[(size_t)(3 * 16 + m) * G4 + col] = c3[r] + bias;
    }
}

// ============================================================
// Kernel 2: sequential 2-layer LSTM (persistent, grid-barrier)
//   64 blocks x 256 threads. Block b owns hidden indices [16b,16b+16).
// ============================================================
__global__ __launch_bounds__(256) void lstm_seq(const float* __restrict__ Wih,
                                                const float* __restrict__ Whh,
                                                const float* __restrict__ bih,
                                                const float* __restrict__ bhh,
                                                float* __restrict__ ws,
                                                float* __restrict__ hT,
                                                float* __restrict__ cT) {
    float* g0   = ws + OFF_G0;
    float* outs = ws + OFF_OUTS;
    float* h0b  = ws + OFF_H0B;
    float* h1b  = ws + OFF_H1B;
    float* c0v  = ws + OFF_C0;
    float* c1v  = ws + OFF_C1;
    unsigned* cnt = (unsigned*)(ws + OFF_CNT);

    const float* Wih1 = Wih + (size_t)G4 * H_DIM;   // layer 1
    const float* Whh0 = Whh;
    const float* Whh1 = Whh + (size_t)G4 * H_DIM;
    const float* bih1 = bih + G4;
    const float* bhh1 = bhh + G4;

    const int tid  = threadIdx.x;
    const int j0   = blockIdx.x * 16;
    const int rl   = tid >> 2;         // 0..63 local row
    const int p    = tid & 3;          // K-quarter
    const int gsel = rl >> 4;          // gate 0..3 (i,f,g,o)
    const int jl   = rl & 15;
    const int row  = gsel * H_DIM + j0 + jl;

    __shared__ __align__(16) float hbuf[2 * H_DIM];
    __shared__ float part[256];
    __shared__ float gv[64];

    for (int t = 0; t < S_LEN; ++t) {
        const float* h0prev = h0b + (t & 1) * H_DIM;
        float*       h0next = h0b + ((t + 1) & 1) * H_DIM;
        const float* h1prev = h1b + (t & 1) * H_DIM;
        float*       h1next = h1b + ((t + 1) & 1) * H_DIM;

        // -------- layer 0: gates = g0[t] + h0prev @ Whh0^T --------
        for (int i = tid; i < H_DIM; i += 256) hbuf[i] = h0prev[i];
        __syncthreads();
        {
            const float4* w4 = (const float4*)(Whh0 + (size_t)row * H_DIM + p * 256);
            const float4* h4 = (const float4*)(hbuf + p * 256);
            float acc = 0.f;
#pragma unroll 8
            for (int kk = 0; kk < 64; ++kk) {
                float4 a = w4[kk], b = h4[kk];
                acc += a.x * b.x + a.y * b.y + a.z * b.z + a.w * b.w;
            }
            part[tid] = acc;
        }
        __syncthreads();
        if (tid < 64) {
            gv[tid] = part[4 * tid] + part[4 * tid + 1] + part[4 * tid + 2] + part[4 * tid + 3]
                    + g0[(size_t)t * G4 + (tid >> 4) * H_DIM + j0 + (tid & 15)];
        }
        __syncthreads();
        if (tid < 16) {
            const int jj = j0 + tid;
            const float iv = gv[tid], fv = gv[16 + tid], gg = gv[32 + tid], ov = gv[48 + tid];
            const float cp = c0v[jj];
            const float cn = sigm(fv) * cp + sigm(iv) * tanhf(gg);
            c0v[jj]    = cn;
            h0next[jj] = sigm(ov) * tanhf(cn);
        }
        grid_barrier(cnt + 2 * t, gridDim.x);

        // -------- layer 1: gates = h0next @ Wih1^T + h1prev @ Whh1^T + b --------
        for (int i = tid; i < H_DIM; i += 256) {
            hbuf[i]         = h0next[i];
            hbuf[H_DIM + i] = h1prev[i];
        }
        __syncthreads();
        {
            const float* wbase = (p < 2) ? (Wih1 + (size_t)row * H_DIM + p * 512)
                                         : (Whh1 + (size_t)row * H_DIM + (p - 2) * 512);
            const float4* w4 = (const float4*)wbase;
            const float4* h4 = (const float4*)(hbuf + p * 512);
            float acc = 0.f;
#pragma unroll 8
            for (int kk = 0; kk < 128; ++kk) {
                float4 a = w4[kk], b = h4[kk];
                acc += a.x * b.x + a.y * b.y + a.z * b.z + a.w * b.w;
            }
            part[tid] = acc;
        }
        __syncthreads();
        if (tid < 64) {
            const int rr = (tid >> 4) * H_DIM + j0 + (tid & 15);
            gv[tid] = part[4 * tid] + part[4 * tid + 1] + part[4 * tid + 2] + part[4 * tid + 3]
                    + bih1[rr] + bhh1[rr];
        }
        __syncthreads();
        if (tid < 16) {
            const int jj = j0 + tid;
            const float iv = gv[tid], fv = gv[16 + tid], gg = gv[32 + tid], ov = gv[48 + tid];
            const float cp = c1v[jj];
            const float cn = sigm(fv) * cp + sigm(iv) * tanhf(gg);
            c1v[jj]    = cn;
            const float hn = sigm(ov) * tanhf(cn);
            h1next[jj] = hn;
            outs[(size_t)t * H_DIM + jj] = hn;
        }
        grid_barrier(cnt + 2 * t + 1, gridDim.x);
    }

    // final states (buffers 0 hold step-64 values: (63+1)&1 == 0)
    if (tid < 16) {
        const int jj = j0 + tid;
        hT[jj]         = h0b[jj];
        hT[H_DIM + jj] = h1b[jj];
        cT[jj]         = c0v[jj];
        cT[H_DIM + jj] = c1v[jj];
    }
}

// ============================================================
// Kernel 3: dot attention. One block per query row s.
// ============================================================
__global__ __launch_bounds__(256) void attention_kernel(const float* __restrict__ outs,
                                                        const float* __restrict__ enc,
                                                        float* __restrict__ attn) {
    const int s = blockIdx.x, tid = threadIdx.x;
    __shared__ __align__(16) float orow[H_DIM];
    __shared__ float red[256];
    __shared__ float sc[LENC];
    __shared__ float inv_sum;

    for (int i = tid; i < H_DIM; i += 256) orow[i] = outs[(size_t)s * H_DIM + i];
    __syncthreads();

    const int l = tid >> 2, p = tid & 3;
    {
        const float4* e4 = (const float4*)(enc + (size_t)l * H_DIM + p * 256);
        const float4* o4 = (const float4*)(orow + p * 256);
        float acc = 0.f;
#pragma unroll 8
        for (int kk = 0; kk < 64; ++kk) {
            float4 a = e4[kk], b = o4[kk];
            acc += a.x * b.x + a.y * b.y + a.z * b.z + a.w * b.w;
        }
        red[tid] = acc;
    }
    __syncthreads();
    if (tid < LENC) sc[tid] = red[4 * tid] + red[4 * tid + 1] + red[4 * tid + 2] + red[4 * tid + 3];
    __syncthreads();
    if (tid == 0) {
        float m = sc[0];
        for (int i = 1; i < LENC; ++i) m = fmaxf(m, sc[i]);
        float ssum = 0.f;
        for (int i = 0; i < LENC; ++i) { float e = __expf(sc[i] - m); sc[i] = e; ssum += e; }
        inv_sum = 1.f / ssum;
    }
    __syncthreads();
    const float inv = inv_sum;

    const int h = tid * 4;
    float4 acc = {0.f, 0.f, 0.f, 0.f};
    for (int ll = 0; ll < LENC; ++ll) {
        const float w = sc[ll] * inv;
        const float4 e = *(const float4*)(enc + (size_t)ll * H_DIM + h);
        acc.x += w * e.x; acc.y += w * e.y; acc.z += w * e.z; acc.w += w * e.w;
    }
    for (int i = tid; i < H_DIM; i += 256) attn[(size_t)s * 2048 + i] = orow[i];
    *(float4*)(attn + (size_t)s * 2048 + H_DIM + h) = acc;
}

// ============================================================
// Kernel 4: logits = outs @ W_fc^T + b_fc via fp32 WMMA
//   * outs (64x1024) staged into LDS (row pitch 1028 -> no bank
//     conflicts) via CDNA5 async global->LDS loads, waited with
//     s_wait_asynccnt.
//   * each wave: 2 N-tiles x 4 M-tiles (8 accumulators), block of
//     8 waves covers 16 N-tiles. grid: 125 blocks x 256.
//   * W_fc (131 MB) streamed from HBM exactly once; A from LDS.
// ============================================================
__global__ __launch_bounds__(256) void logits_wmma(const float* __restrict__ outs,
                                                   const float* __restrict__ Wfc,
                                                   const float* __restrict__ bfc,
                                                   float* __restrict__ logits) {
    extern __shared__ float smem[];                 // 64 * ROWP floats

    const int tid = threadIdx.x;

    // ---- stage outs -> LDS with async b128 copies (ASYNCcnt) ----
    for (int i = tid; i < 64 * 256; i += 256) {     // 16384 x 16B transfers
        const int r  = i >> 8;                      // row 0..63
        const int c4 = (i & 255) * 4;               // col 0..1020 step 4
        unsigned lds_addr = (unsigned)(uintptr_t)(smem + r * ROWP + c4);
        unsigned long long gaddr =
            (unsigned long long)(uintptr_t)(outs + (size_t)r * H_DIM + c4);
        asm volatile("global_load_async_to_lds_b128 %0, %1, off"
                     :: "v"(lds_addr), "v"(gaddr) : "memory");
    }
    asm volatile("s_wait_asynccnt 0x0" ::: "memory");
    __syncthreads();

    const int wv   = tid >> 5;                      // 0..7
    const int lane = tid & 31;
    const int half = lane >> 4;
    const int l15  = lane & 15;
    const int nt0  = (blockIdx.x * 8 + wv) * 2;     // first of 2 N tiles
    const int colA = nt0 * 16 + l15;
    const int colB = colA + 16;

    v8f cA0 = {}, cA1 = {}, cA2 = {}, cA3 = {};
    v8f cB0 = {}, cB1 = {}, cB2 = {}, cB3 = {};
    for (int k = 0; k < H_DIM; k += 4) {
        const int kb = k + half * 2;
        v2f bA = *(const v2f*)(Wfc + (size_t)colA * H_DIM + kb);
        v2f bB = *(const v2f*)(Wfc + (size_t)colB * H_DIM + kb);
        v2f a0 = *(const v2f*)(smem + (0 * 16 + l15) * ROWP + kb);
        v2f a1 = *(const v2f*)(smem + (1 * 16 + l15) * ROWP + kb);
        v2f a2 = *(const v2f*)(smem + (2 * 16 + l15) * ROWP + kb);
        v2f a3 = *(const v2f*)(smem + (3 * 16 + l15) * ROWP + kb);
        cA0 = __builtin_amdgcn_wmma_f32_16x16x4_f32(false, a0, false, bA, (short)0, cA0, false, false);
        cA1 = __builtin_amdgcn_wmma_f32_16x16x4_f32(false, a1, false, bA, (short)0, cA1, false, false);
        cA2 = __builtin_amdgcn_wmma_f32_16x16x4_f32(false, a2, false, bA, (short)0, cA2, false, false);
        cA3 = __builtin_amdgcn_wmma_f32_16x16x4_f32(false, a3, false, bA, (short)0, cA3, false, false);
        cB0 = __builtin_amdgcn_wmma_f32_16x16x4_f32(false, a0, false, bB, (short)0, cB0, false, false);
        cB1 = __builtin_amdgcn_wmma_f32_16x16x4_f32(false, a1, false, bB, (short)0, cB1, false, false);
        cB2 = __builtin_amdgcn_wmma_f32_16x16x4_f32(false, a2, false, bB, (short)0, cB2, false, false);
        cB3 = __builtin_amdgcn_wmma_f32_16x16x4_f32(false, a3, false, bB, (short)0, cB3, false, false);
    }

    const float biasA = bfc[colA];
    const float biasB = bfc[colB];
#pragma unroll
    for (int r = 0; r < 8; ++r) {
        const int m = r + half * 8;
        logits[(size_t)(0 * 16 + m) * V_DIM + colA] = cA0[r] + biasA;
        logits[(size_t)(1 * 16 + m) * V_DIM + colA] = cA1[r] + biasA;
        logits[(size_t)(2 * 16 + m) * V_DIM + colA] = cA2[r] + biasA;
        logits[(size_t)(3 * 16 + m) * V_DIM + colA] = cA3[r] + biasA;
        logits[(size_t)(0 * 16 + m) * V_DIM + colB] = cB0[r] + biasB;
        logits[(size_t)(1 * 16 + m) * V_DIM + colB] = cB1[r] + biasB;
        logits[(size_t)(2 * 16 + m) * V_DIM + colB] = cB2[r] + biasB;
        logits[(size_t)(3 * 16 + m) * V_DIM + colB] = cB3[r] + biasB;
    }
}

// ============================================================
extern "C" void kernel_launch(void* const* d_in, const int* in_sizes, int n_in,
                              void* d_out, int out_size, void* d_ws, size_t ws_size,
                              hipStream_t stream) {
    const float* dec  = (const float*)d_in[0];   // [1,64,512]
    const float* z    = (const float*)d_in[1];   // [1,512]
    const float* enc  = (const float*)d_in[2];   // [1,64,1024]
    const float* h0   = (const float*)d_in[3];   // [2,1,1024]
    const float* c0   = (const float*)d_in[4];   // [2,1,1024]
    const float* Wih  = (const float*)d_in[5];   // [2,4096,1024]
    const float* Whh  = (const float*)d_in[6];   // [2,4096,1024]
    const float* bih  = (const float*)d_in[7];   // [2,4096]
    const float* bhh  = (const float*)d_in[8];   // [2,4096]
    const float* Wfc  = (const float*)d_in[9];   // [32000,1024]
    const float* bfc  = (const float*)d_in[10];  // [32000]
    (void)d_in[11]; (void)in_sizes; (void)n_in; (void)out_size; (void)ws_size;

    float* ws  = (float*)d_ws;
    float* out = (float*)d_out;

    // allow >64KB dynamic LDS for the staging kernel (no-op if unneeded)
    static bool attr_set = false;
    if (!attr_set) {
        (void)hipFuncSetAttribute((const void*)logits_wmma,
                                  hipFuncAttributeMaxDynamicSharedMemorySize, SMEM_BYTES);
        attr_set = true;
    }

    init_state<<<1, 1024, 0, stream>>>(h0, c0, ws);
    g0_wmma<<<32, 256, 0, stream>>>(dec, z, Wih /*layer0*/, bih, bhh, ws + OFF_G0);
    lstm_seq<<<64, 256, 0, stream>>>(Wih, Whh, bih, bhh, ws,
                                     out + OUT_HT, out + OUT_CT);
    attention_kernel<<<64, 256, 0, stream>>>(ws + OFF_OUTS, enc, out + OUT_ATTN);
    logits_wmma<<<125, 256, SMEM_BYTES, stream>>>(ws + OFF_OUTS, Wfc, bfc, out + OUT_LOGITS);
}